// SAE_36275293782067
// MI455X (gfx1250) — compile-verified
//
#include <hip/hip_runtime.h>
#include <hip/hip_bf16.h>

typedef float v2f __attribute__((ext_vector_type(2)));
typedef float v8f __attribute__((ext_vector_type(8)));

#define B_ROWS 8192
#define D_IN   1024
#define D_LAT  16384
#define TOPK   64

#define BM 128          // block tile M
#define BN 128          // block tile N
#define KT 32           // K step per stage
#define LSTR 36         // LDS row stride (floats): conflict-free + 16B aligned rows

// ---------------------------------------------------------------------------
// Encode: h = relu(x @ W_enc^T + b_enc)
// Block tile 128x128, K-step 32, double-buffered LDS (72 KB of 320 KB WGP LDS).
// 256 threads = 8 wave32s in 4(M) x 2(N); wave tile 32x64 = 8 accumulators of
// V_WMMA_F32_16X16X4_F32 -> B fragments reused across 2 M sub-tiles.
// ---------------------------------------------------------------------------
__global__ __launch_bounds__(256)
void sae_encode_wmma(const float* __restrict__ x,
                     const float* __restrict__ Wenc,
                     const float* __restrict__ benc,
                     float* __restrict__ h)
{
    __shared__ float xs [2][BM][LSTR];
    __shared__ float wsh[2][BN][LSTR];

    const int tid  = threadIdx.x;
    const int wave = tid >> 5;
    const int lane = tid & 31;
    const int half = lane >> 4;        // 0: lanes 0-15, 1: lanes 16-31
    const int l15  = lane & 15;
    const int wm   = (wave & 3) * 32;  // wave M offset in block
    const int wn   = (wave >> 2) * 64; // wave N offset in block

    const int blockN = blockIdx.x * BN;
    const int blockM = blockIdx.y * BM;

    v8f acc[2][4];
    #pragma unroll
    for (int mt = 0; mt < 2; ++mt)
        #pragma unroll
        for (int t = 0; t < 4; ++t) acc[mt][t] = (v8f){};

    // register staging: 4 float4 of x-tile + 4 float4 of W-tile per thread
    float4 rx[4], rw[4];

    // prologue: fetch K-stage 0
    #pragma unroll
    for (int i = 0; i < 4; ++i) {
        const int f4 = tid + i * 256;          // 0..1023
        const int r  = f4 >> 3;
        const int c  = (f4 & 7) << 2;
        rx[i] = *(const float4*)(x    + (size_t)(blockM + r) * D_IN + c);
        rw[i] = *(const float4*)(Wenc + (size_t)(blockN + r) * D_IN + c);
    }
    #pragma unroll
    for (int i = 0; i < 4; ++i) {
        const int f4 = tid + i * 256;
        const int r  = f4 >> 3;
        const int c  = (f4 & 7) << 2;
        *(float4*)&xs [0][r][c] = rx[i];
        *(float4*)&wsh[0][r][c] = rw[i];
    }
    __syncthreads();

    const int NSTAGE = D_IN / KT;              // 32
    for (int s = 0; s < NSTAGE; ++s) {
        const int cur = s & 1;

        // issue next-stage global loads before the WMMA burst
        if (s + 1 < NSTAGE) {
            const int kk = (s + 1) * KT;
            #pragma unroll
            for (int i = 0; i < 4; ++i) {
                const int f4 = tid + i * 256;
                const int r  = f4 >> 3;
                const int c  = (f4 & 7) << 2;
                rx[i] = *(const float4*)(x    + (size_t)(blockM + r) * D_IN + kk + c);
                rw[i] = *(const float4*)(Wenc + (size_t)(blockN + r) * D_IN + kk + c);
            }
        }

        // compute current stage
        #pragma unroll
        for (int k4 = 0; k4 < 8; ++k4) {
            // A 16x4 f32 layout: VGPR0 = K0 (lanes<16)/K2 (lanes>=16); VGPR1 = K1/K3
            const int kc = k4 * 4 + half * 2;
            v2f a0, a1;
            a0.x = xs[cur][wm +      l15][kc + 0];
            a0.y = xs[cur][wm +      l15][kc + 1];
            a1.x = xs[cur][wm + 16 + l15][kc + 0];
            a1.y = xs[cur][wm + 16 + l15][kc + 1];
            #pragma unroll
            for (int t = 0; t < 4; ++t) {
                v2f b;
                b.x = wsh[cur][wn + t * 16 + l15][kc + 0];
                b.y = wsh[cur][wn + t * 16 + l15][kc + 1];
                acc[0][t] = __builtin_amdgcn_wmma_f32_16x16x4_f32(
                    false, a0, false, b, (short)0, acc[0][t], false, false);
                acc[1][t] = __builtin_amdgcn_wmma_f32_16x16x4_f32(
                    false, a1, false, b, (short)0, acc[1][t], false, false);
            }
        }

        // retire staged regs into the other buffer
        if (s + 1 < NSTAGE) {
            const int nxt = cur ^ 1;
            #pragma unroll
            for (int i = 0; i < 4; ++i) {
                const int f4 = tid + i * 256;
                const int r  = f4 >> 3;
                const int c  = (f4 & 7) << 2;
                *(float4*)&xs [nxt][r][c] = rx[i];
                *(float4*)&wsh[nxt][r][c] = rw[i];
            }
        }
        __syncthreads();
    }

    // epilogue: bias + relu. C/D layout: VGPR r, lane -> M = r + 8*half, N = l15.
    #pragma unroll
    for (int t = 0; t < 4; ++t) {
        const int col   = blockN + wn + t * 16 + l15;
        const float bia = benc[col];
        #pragma unroll
        for (int mt = 0; mt < 2; ++mt) {
            #pragma unroll
            for (int r = 0; r < 8; ++r) {
                const int m = blockM + wm + mt * 16 + half * 8 + r;
                float v = acc[mt][t][r] + bia;
                v = v > 0.0f ? v : 0.0f;
                h[(size_t)m * D_LAT + col] = v;
            }
        }
    }
}

// ---------------------------------------------------------------------------
// Exact top-64 per row via 4-round radix select on float bit patterns
// (valid: h >= 0 after relu, so uint compare == float compare).
// Row (64 KB) cached in LDS once: radix rounds + selection read LDS, not global.
// Produces h_topk and an index-sorted (val, idx) list -> deterministic decode.
// ---------------------------------------------------------------------------
__global__ __launch_bounds__(256)
void sae_topk(const float* __restrict__ h,
              float* __restrict__ h_topk,
              float* __restrict__ cvals,
              int*   __restrict__ cidx)
{
    const int row = blockIdx.x;
    const int tid = threadIdx.x;
    const float* hr = h + (size_t)row * D_LAT;

    __shared__ float    rowbuf[D_LAT];     // 64 KB row cache
    __shared__ int      hist[256];
    __shared__ unsigned s_prefix;
    __shared__ int      s_krem;
    __shared__ int      s_cnt, s_cnt2;
    __shared__ float    sv[64];
    __shared__ int      si[64];
    __shared__ int      eqi[64];

    // pull the row into LDS (float4 loads, coalesced)
    #pragma unroll
    for (int i = 0; i < D_LAT / (256 * 4); ++i) {
        const int e = (tid + i * 256) * 4;
        const float4 v = *(const float4*)(hr + e);
        *(float4*)&rowbuf[e] = v;
    }
    if (tid == 0) { s_prefix = 0u; s_krem = TOPK; }

    for (int round = 0; round < 4; ++round) {
        hist[tid] = 0;
        __syncthreads();
        const unsigned prefix = s_prefix;
        const int krem = s_krem;
        const int shift = 24 - 8 * round;
        for (int i = tid; i < D_LAT; i += 256) {
            unsigned u = __float_as_uint(rowbuf[i]);
            bool match = (round == 0) || ((u >> (shift + 8)) == prefix);
            if (match) atomicAdd(&hist[(u >> shift) & 0xFF], 1);
        }
        __syncthreads();
        if (tid == 0) {
            int acc = 0, d = 255;
            for (; d > 0; --d) {
                if (acc + hist[d] >= krem) break;
                acc += hist[d];
            }
            s_krem   = krem - acc;
            s_prefix = (prefix << 8) | (unsigned)d;
        }
        __syncthreads();
    }

    const unsigned T = s_prefix;       // bit pattern of the k-th largest value
    const int kEq = s_krem;            // how many ==T entries to include
    const int kGt = TOPK - kEq;        // count of entries strictly > T

    if (tid == 0) { s_cnt = 0; s_cnt2 = 0; }
    __syncthreads();

    float* htr = h_topk + (size_t)row * D_LAT;
    for (int i = tid; i < D_LAT; i += 256) {
        float v = rowbuf[i];
        unsigned u = __float_as_uint(v);
        bool gt = (u > T);
        htr[i] = gt ? v : 0.0f;        // ==T entries patched below
        if (gt) {
            int slot = atomicAdd(&s_cnt, 1);
            if (slot < TOPK) { sv[slot] = v; si[slot] = i; }
        }
        if (T != 0u && u == T) {
            int slot = atomicAdd(&s_cnt2, 1);
            if (slot < 64) eqi[slot] = i;
        }
    }
    __syncthreads();

    if (tid == 0) {
        if (T != 0u) {
            int m = s_cnt2; if (m > 64) m = 64;
            // insertion sort tie indices ascending (stable, matches lax.top_k)
            for (int a = 1; a < m; ++a) {
                int key = eqi[a]; int b = a - 1;
                while (b >= 0 && eqi[b] > key) { eqi[b + 1] = eqi[b]; --b; }
                eqi[b + 1] = key;
            }
            for (int j = 0; j < kEq; ++j) {
                si[kGt + j] = eqi[j];
                sv[kGt + j] = __uint_as_float(T);
            }
        } else {
            // threshold is 0.0: selected zeros are decode-neutral; pad.
            for (int j = 0; j < kEq; ++j) { si[kGt + j] = 0; sv[kGt + j] = 0.0f; }
        }
    }
    __syncthreads();

    if (T != 0u && tid < kEq) htr[si[kGt + tid]] = __uint_as_float(T);

    // bitonic sort 64 (val,idx) pairs by idx ascending -> deterministic order
    for (int ksz = 2; ksz <= 64; ksz <<= 1) {
        for (int j = ksz >> 1; j > 0; j >>= 1) {
            __syncthreads();
            if (tid < 64) {
                int ixj = tid ^ j;
                if (ixj > tid) {
                    bool ascend = ((tid & ksz) == 0);
                    int ia = si[tid], ib = si[ixj];
                    bool doswap = ascend ? (ia > ib) : (ia < ib);
                    if (doswap) {
                        si[tid] = ib; si[ixj] = ia;
                        float fa = sv[tid]; sv[tid] = sv[ixj]; sv[ixj] = fa;
                    }
                }
            }
        }
    }
    __syncthreads();
    if (tid < TOPK) {
        cvals[(size_t)row * TOPK + tid] = sv[tid];
        cidx [(size_t)row * TOPK + tid] = si[tid];
    }
}

// ---------------------------------------------------------------------------
// Sparse decode: x_hat[b,:] = b_dec + sum_{j<64} val_j * W_dec[:, idx_j]
// One block per row; W_dec (64 MB) is L2-resident on MI455X (192 MB L2).
// Fixed ascending-index accumulation order -> deterministic output.
// ---------------------------------------------------------------------------
__global__ __launch_bounds__(256)
void sae_decode(const float* __restrict__ Wdec,
                const float* __restrict__ bdec,
                const float* __restrict__ cvals,
                const int*   __restrict__ cidx,
                float* __restrict__ xhat)
{
    const int row = blockIdx.x;
    const int tid = threadIdx.x;
    __shared__ float sv[64];
    __shared__ int   si[64];
    if (tid < 64) {
        sv[tid] = cvals[(size_t)row * TOPK + tid];
        si[tid] = cidx [(size_t)row * TOPK + tid];
    }
    __syncthreads();

    #pragma unroll
    for (int j = 0; j < D_IN / 256; ++j) {
        const int d = tid + j * 256;
        const float* wr = Wdec + (size_t)d * D_LAT;
        float acc = bdec[d];
        #pragma unroll 8
        for (int l = 0; l < TOPK; ++l) acc += sv[l] * wr[si[l]];
        xhat[(size_t)row * D_IN + d] = acc;
    }
}

extern "C" void kernel_launch(void* const* d_in, const int* in_sizes, int n_in,
                              void* d_out, int out_size, void* d_ws, size_t ws_size,
                              hipStream_t stream)
{
    const float* x    = (const float*)d_in[0];
    const float* Wenc = (const float*)d_in[1];
    const float* benc = (const float*)d_in[2];
    const float* Wdec = (const float*)d_in[3];
    const float* bdec = (const float*)d_in[4];
    // d_in[5] is k; fixed to 64 per the reference setup.

    float* xhat   = (float*)d_out;
    float* h      = xhat + (size_t)B_ROWS * D_IN;
    float* h_topk = h    + (size_t)B_ROWS * D_LAT;

    float* cvals = (float*)d_ws;                                                // 2 MB
    int*   cidx  = (int*)((char*)d_ws + (size_t)B_ROWS * TOPK * sizeof(float)); // 2 MB

    dim3 egrid(D_LAT / BN, B_ROWS / BM);   // 128 x 64
    sae_encode_wmma<<<egrid, 256, 0, stream>>>(x, Wenc, benc, h);
    sae_topk<<<B_ROWS, 256, 0, stream>>>(h, h_topk, cvals, cidx);
    sae_decode<<<B_ROWS, 256, 0, stream>>>(Wdec, bdec, cvals, cidx, xhat);
}